// GeekMeshRasterizer_2774548873455
// MI455X (gfx1250) — compile-verified
//
#include <hip/hip_runtime.h>
#include <stdint.h>

// Problem constants (match reference)
#define HH   256
#define WW   256
#define KK   8
#define NF   5000
#define BGZ  1.1f
#define EPSA 1e-8f

// Face record: 24 floats (96B), all WMMA A-operand pairs 8B-aligned:
// [0..5]   = (A0,B0)(A1,B1)(A2,B2)  edge coefficients, sign-flipped so area>0
// [6..11]  = (C0,0)(C1,0)(C2,0)     edge constants as aligned pairs
// [12]     = 1/|area|  (NaN if invalid -> auto-reject)
// [13]     = 0
// [14..19] = v0x,v0y,v1x,v1y,v2x,v2y (scaled, unflipped) for segment distances
// [20..23] = 0 padding
#define RECF 24
#define CHUNK  256                 // faces per LDS chunk (24KB)
#define NCHUNK 20                  // 20*256 = 5120 >= 5000
#define FPAD   (CHUNK * NCHUNK)
#define CHUNKF (CHUNK * RECF)      // floats per chunk

typedef float v2f  __attribute__((ext_vector_type(2)));
typedef float v8f  __attribute__((ext_vector_type(8)));
typedef unsigned int v4u __attribute__((ext_vector_type(4)));
typedef int   v4i  __attribute__((ext_vector_type(4)));
typedef int   v8i  __attribute__((ext_vector_type(8)));

// ---------------------------------------------------------------- prep kernel
__global__ __launch_bounds__(256) void prep_faces(const float* __restrict__ verts,
                                                  const int* __restrict__ faces,
                                                  float* __restrict__ rec) {
  int f = blockIdx.x * blockDim.x + threadIdx.x;
  if (f >= FPAD) return;
  float* r = rec + (size_t)f * RECF;
#pragma unroll
  for (int i = 0; i < RECF; ++i) r[i] = 0.0f;
  if (f >= NF) {                       // padding faces: reject via NaN inv
    r[12] = __builtin_nanf("");
    return;
  }
  const float sx = (float)WW / (float)HH;   // == 1.0 here, kept general
  int i0 = faces[f * 3 + 0], i1 = faces[f * 3 + 1], i2 = faces[f * 3 + 2];
  float v0x = verts[i0 * 3 + 0] * sx, v0y = verts[i0 * 3 + 1];
  float v1x = verts[i1 * 3 + 0] * sx, v1y = verts[i1 * 3 + 1];
  float v2x = verts[i2 * 3 + 0] * sx, v2y = verts[i2 * 3 + 1];
  // w0 = (v2x-v1x)(py-v1y) - (v2y-v1y)(px-v1x) = A0*px + B0*py + C0, etc.
  float A0 = v1y - v2y, B0 = v2x - v1x, C0 = (v2y - v1y) * v1x - (v2x - v1x) * v1y;
  float A1 = v2y - v0y, B1 = v0x - v2x, C1 = (v0y - v2y) * v2x - (v0x - v2x) * v2y;
  float A2 = v0y - v1y, B2 = v1x - v0x, C2 = (v1y - v0y) * v0x - (v1x - v0x) * v0y;
  float area = (v1x - v0x) * (v2y - v0y) - (v1y - v0y) * (v2x - v0x);
  bool  valid = fabsf(area) > EPSA;
  // Flip so effective area > 0: w' = s*w, inv' = 1/(s*area); b = w'*inv' unchanged.
  float s = (area < 0.0f) ? -1.0f : 1.0f;
  float inv = valid ? (1.0f / (s * area)) : __builtin_nanf("");
  r[0] = s * A0; r[1] = s * B0; r[2] = s * A1; r[3] = s * B1; r[4] = s * A2; r[5] = s * B2;
  r[6] = s * C0; r[8] = s * C1; r[10] = s * C2;   // odd slots stay 0
  r[12] = inv;
  r[14] = v0x; r[15] = v0y; r[16] = v1x; r[17] = v1y; r[18] = v2x; r[19] = v2y;
}

// ------------------------------------------------------------- TDM staging
#if __has_builtin(__builtin_amdgcn_tensor_load_to_lds)
#define USE_TDM 1
__device__ __forceinline__ void tdm_load_chunk(const float* gsrc, uint32_t ldsByteOff) {
  uint64_t ga = (uint64_t)(uintptr_t)gsrc;
  // D# group 0: count=1, lds_addr, global_addr[56:0], type=2
  v4u g0;
  g0.x = 1u;
  g0.y = ldsByteOff;
  g0.z = (uint32_t)(ga & 0xFFFFFFFFu);
  g0.w = (uint32_t)((ga >> 32) & 0x01FFFFFFu) | (2u << 30);
  // D# group 1: data_size=4B, 1-D tensor: tensor_dim0 = tile_dim0 = CHUNKF floats
  const uint32_t td0 = (uint32_t)CHUNKF, tile0 = (uint32_t)CHUNKF;
  v8i g1;
  g1[0] = (int)(2u << 16);                                   // data_size=2 (4B)
  g1[1] = (int)((td0 & 0xFFFFu) << 16);                      // tensor_dim0[15:0]
  g1[2] = (int)(((td0 >> 16) & 0xFFFFu) | (1u << 16));       // tensor_dim0[31:16], tensor_dim1=1
  g1[3] = (int)((tile0 & 0xFFFFu) << 16);                    // tile_dim0
  g1[4] = 0;                                                 // tile_dim1/2 unused
  g1[5] = (int)td0;                                          // tensor_dim0_stride[31:0]
  g1[6] = 0;
  g1[7] = 0;
  v4i z4 = {0, 0, 0, 0};
  v8i z8 = {0, 0, 0, 0, 0, 0, 0, 0};
  __builtin_amdgcn_tensor_load_to_lds(g0, g1, z4, z4, z8, 0);
}
__device__ __forceinline__ void wait_tensor0() {
#if __has_builtin(__builtin_amdgcn_s_wait_tensorcnt)
  __builtin_amdgcn_s_wait_tensorcnt(0);
#else
  asm volatile("s_wait_tensorcnt 0x0" ::: "memory");
#endif
}
#else
#define USE_TDM 0
#endif

__device__ __forceinline__ void coop_copy(float* dst, const float* src) {
  for (int i = threadIdx.x; i < CHUNKF / 4; i += blockDim.x)
    ((float4*)dst)[i] = ((const float4*)src)[i];
}

// ---------------------------------------------------------------- helpers
__device__ __forceinline__ float seg_d2(float ax, float ay, float bx, float by,
                                        float px, float py) {
  float dx = bx - ax, dy = by - ay;
  float l2 = fmaxf(fmaf(dx, dx, dy * dy), 1e-12f);
  float t  = (fmaf(px - ax, dx, (py - ay) * dy)) / l2;
  t = fminf(fmaxf(t, 0.0f), 1.0f);
  float qx = fmaf(t, dx, ax), qy = fmaf(t, dy, ay);
  float ex = px - qx, ey = py - qy;
  return fmaf(ex, ex, ey * ey);
}

// area flipped positive -> inside test is min3(w) >= 0; invalid faces have
// inv = NaN -> z = NaN -> key never accepted (NaN < x is false).
__device__ __forceinline__ void topk_try(float w0, float w1, float w2, float inv,
                                         int fidx, float zk[KK], int fk[KK]) {
  float mn = fminf(fminf(w0, w1), w2);            // v_min3_f32
  float z  = ((w0 + w1 + w2) * inv) * BGZ;
  float key = (mn >= 0.0f) ? z : __builtin_inff();
  if (key < zk[KK - 1]) {                         // rare: wave-branch
    float cz = key; int ci = fidx;
#pragma unroll
    for (int s = 0; s < KK; ++s) {                // stable insertion (strict <)
      bool repl = cz < zk[s];
      float tz = zk[s]; int ti = fk[s];
      zk[s] = repl ? cz : tz;  fk[s] = repl ? ci : ti;
      cz   = repl ? tz : cz;   ci   = repl ? ti : ci;
    }
  }
}

// ---------------------------------------------------------------- main kernel
__global__ __launch_bounds__(256) void raster_main(const float* __restrict__ rec,
                                                   int* __restrict__ out_p2f,
                                                   float* __restrict__ out_zb,
                                                   float* __restrict__ out_bg,
                                                   float* __restrict__ out_dg) {
  __shared__ float smem[2 * CHUNKF];                    // 48KB double buffer
  const int tid  = threadIdx.x;
  const int wave = tid >> 5;
  const int lane = tid & 31;
  const int lp   = lane & 15;                           // pixel within 16-wide strip
  const bool hi  = lane >= 16;
  const int x = blockIdx.x * 16 + lp;
  const int y = blockIdx.y * 8 + wave;
  const float px = 1.0f - (2.0f * (float)x + 1.0f) / (float)WW;
  const float py = 1.0f - (2.0f * (float)y + 1.0f) / (float)HH;

  // B matrix (4x16): rows (px,py,1,0); lanes 0-15 supply K=0,1; lanes 16-31 K=2,3
  v2f bmat;
  bmat.x = hi ? 1.0f : px;
  bmat.y = hi ? 0.0f : py;

  // A-matrix row this lane supplies: row = lane&15 = 4*face_in_group + edge.
  // Branchless loop-invariant float offset into the face record:
  //   low lanes e<3 : (A_e,B_e) at 2e ; hi lanes e<3 : (C_e,0) at 6+2e
  //   e==3 rows are don't-care (D rows 3/7/11/15 never read) -> read offset 0.
  const int fsel = lp >> 2;
  const int e    = lp & 3;
  const int ofsA = (e < 3) ? (hi ? (6 + 2 * e) : (2 * e)) : 0;
  const int finv = hi ? 2 : 0;                          // face-in-group of d[0..2]

  float zk[KK]; int fk[KK];
#pragma unroll
  for (int s = 0; s < KK; ++s) { zk[s] = __builtin_inff(); fk[s] = -1; }

#if USE_TDM
  if (wave == 0) tdm_load_chunk(rec, (uint32_t)(uintptr_t)(&smem[0]));
#endif

  for (int c = 0; c < NCHUNK; ++c) {
    float* buf = smem + (c & 1) * CHUNKF;
#if USE_TDM
    if (wave == 0) wait_tensor0();                      // chunk c resident in LDS
    __syncthreads();
    if (wave == 0 && (c + 1) < NCHUNK)                  // pipeline next chunk
      tdm_load_chunk(rec + (size_t)(c + 1) * CHUNKF,
                     (uint32_t)(uintptr_t)(&smem[((c + 1) & 1) * CHUNKF]));
#else
    coop_copy(buf, rec + (size_t)c * CHUNKF);
    __syncthreads();
#endif
    const int fbase = c * CHUNK;
#pragma unroll 4
    for (int g = 0; g < CHUNK / 4; ++g) {
      const float* frA = buf + (g * 4 + fsel) * RECF;   // A-operand source row
      v2f a = *(const v2f*)(frA + ofsA);                // one ds_load_b64, no branches
      v8f cacc = {0.f, 0.f, 0.f, 0.f, 0.f, 0.f, 0.f, 0.f};
      // D[row][pix] = A_row*px + B_row*py + C_row  (16 edge rows x 16 pixels)
      v8f d = __builtin_amdgcn_wmma_f32_16x16x4_f32(false, a, false, bmat,
                                                    (short)0, cacc, false, false);
      // this lane's pixel = lp; faces: rows 0-7 (lo lanes) / 8-15 (hi lanes)
      const float* frI = buf + (g * 4 + finv) * RECF;
      const float invA = frI[12];                       // merged ds_load_2addr
      const float invB = frI[12 + RECF];
      const int fA = fbase + g * 4 + finv;
      topk_try(d[0], d[1], d[2], invA, fA,     zk, fk);
      topk_try(d[4], d[5], d[6], invB, fA + 1, zk, fk);
    }
    __syncthreads();                                    // done reading buf before reuse
  }

  // ---- merge the two half-wave lists per pixel through LDS ----
  float* mz = smem;                                     // 128 px * 16 entries
  int*   mi = (int*)(smem + 128 * 16);
  const int p = wave * 16 + lp;
#pragma unroll
  for (int s = 0; s < KK; ++s) {
    int slot = p * 16 + (hi ? (8 + s) : s);
    mz[slot] = zk[s]; mi[slot] = fk[s];
  }
  __syncthreads();

  if (!hi) {
    const int base = p * 16;
    const int q0 = (y * WW + x) * KK;
    int ia = 0, ib = 8;
#pragma unroll
    for (int s = 0; s < KK; ++s) {
      int iaC = ia > 7 ? 7 : ia, ibC = ib > 15 ? 15 : ib;
      float za = mz[base + iaC]; int ja = mi[base + iaC];
      float zb = mz[base + ibC]; int jb = mi[base + ibC];
      bool takeA = (ia <= 7) && ((ib > 15) || (za < zb) || (za == zb && ja < jb));
      float zv = takeA ? za : zb;
      int  fi  = takeA ? ja : jb;
      ia += takeA ? 1 : 0; ib += takeA ? 0 : 1;

      const int q = q0 + s;
      if (fi >= 0) {                                    // hit slot: full outputs
        const float* r = rec + (size_t)fi * RECF;
        float w0 = fmaf(r[0], px, fmaf(r[1], py, r[6]));
        float w1 = fmaf(r[2], px, fmaf(r[3], py, r[8]));
        float w2 = fmaf(r[4], px, fmaf(r[5], py, r[10]));
        float inv = r[12];
        float b0 = w0 * inv, b1 = w1 * inv, b2 = w2 * inv;
        float d01 = seg_d2(r[14], r[15], r[16], r[17], px, py);
        float d12 = seg_d2(r[16], r[17], r[18], r[19], px, py);
        float d20 = seg_d2(r[18], r[19], r[14], r[15], px, py);
        float d2 = fminf(fminf(d01, d12), d20);
        out_p2f[q] = fi;
        out_zb[q]  = zv;
        out_bg[q * 3 + 0] = b0; out_bg[q * 3 + 1] = b1; out_bg[q * 3 + 2] = b2;
        out_dg[q]  = -d2;                               // inside => dist = -d2
      } else {
        out_p2f[q] = -1;
        out_zb[q]  = -1.0f;
        out_bg[q * 3 + 0] = -1.0f; out_bg[q * 3 + 1] = -1.0f; out_bg[q * 3 + 2] = -1.0f;
        out_dg[q]  = -1.0f;
      }
    }
  }
}

// ---------------------------------------------------------------- launcher
extern "C" void kernel_launch(void* const* d_in, const int* in_sizes, int n_in,
                              void* d_out, int out_size, void* d_ws, size_t ws_size,
                              hipStream_t stream) {
  const float* verts = (const float*)d_in[0];   // (3F,3) f32
  const int*   faces = (const int*)d_in[1];     // (F,3) i32
  float* rec = (float*)d_ws;                    // FPAD*24 floats = 480KB scratch

  prep_faces<<<FPAD / 256, 256, 0, stream>>>(verts, faces, rec);

  const int HWK = HH * WW * KK;
  int*   p2f = (int*)d_out;                     // int32 bits in-place
  float* zb  = (float*)d_out + HWK;
  float* bg  = zb + HWK;
  float* dg  = bg + 3 * HWK;

  dim3 grid(WW / 16, HH / 8);                   // 16x8 pixel tile, 8 waves/block
  raster_main<<<grid, 256, 0, stream>>>(rec, p2f, zb, bg, dg);
}